// BlockRadiusMixer_72799695667338
// MI455X (gfx1250) — compile-verified
//
#include <hip/hip_runtime.h>
#include <math.h>

// ---------------- problem constants ----------------
#define BATCH   16384
#define NBLK    32
#define D       128
#define FEAT    (NBLK * D)      // 4096
#define ROWS_WG 128             // rows per workgroup (8 waves x 16 rows)
#define XS      132             // LDS row stride for x tile (bank-conflict-free A reads)
#define SB_FLOATS (256 * 64)    // 16384 floats = 65,536 B, swizzled B fragments
#define SX_FLOATS (ROWS_WG * XS)

typedef __attribute__((ext_vector_type(2))) float v2f;
typedef __attribute__((ext_vector_type(4))) float v4f;
typedef __attribute__((ext_vector_type(8))) float v8f;

// ---------------- pass 0: zero the per-row sum-of-squares accumulator ----------------
__global__ void __launch_bounds__(256) brm_zero_sumsq(float* __restrict__ ws) {
    ws[blockIdx.x * 256 + threadIdx.x] = 0.0f;
}

// ---------------- pass 1: per-block GEMM (f32 WMMA) + ReLU + sumsq atomics ----------------
//
// LDS map (single array so sB is guaranteed at offset 0):
//   smem[0      .. 16383] = sB : 256 segments, seg = t*32 + k, 32 float2 slots each.
//        slot sigma = (2*c16 + half + t) & 31 holds (Q[4k+2*half][t*16+c16],
//                                                    Q[4k+2*half+1][t*16+c16])
//        -> B fragment = ds_load_b64 at sigma*8 (VGPR) + (t*8192 + k*256) immediate
//           (max 65,280 B, fits the 16-bit DS offset; no per-load VALU)
//   smem[16384 ..      ] = sX : x tile, padded row stride XS=132 (conflict-free A reads)
__global__ void __launch_bounds__(256) brm_block_gemm_relu(
    const float* __restrict__ x,      // [BATCH, FEAT]
    const float* __restrict__ Q,      // [NBLK, D, D]
    float* __restrict__ z,            // [BATCH, FEAT]  (d_out, pre-normalize)
    float* __restrict__ sumsq)        // [BATCH]
{
    __shared__ __align__(16) float smem[SB_FLOATS + SX_FLOATS];  // 133,120 B
    float* sB = smem;
    float* sX = smem + SB_FLOATS;

    const int tile  = blockIdx.x;        // row tile index (0..127)
    const int n     = blockIdx.y;        // radius block (0..31)
    const int tid   = threadIdx.x;
    const int row0g = tile * ROWS_WG;

    // ---- stage x tile (coalesced float4 -> padded row-major LDS) ----
    {
        const float* xg = x + (size_t)row0g * FEAT + (size_t)n * D;
        const float* qg = Q + (size_t)n * D * D;
#pragma unroll
        for (int i = 0; i < 16; ++i) {
            const int flatv = i * 256 + tid;      // float4 index, 4096 total
            const int r = flatv >> 5;             // row (32 float4 per 128-col row)
            const int c = (flatv & 31) * 4;
            v4f vx = *(const v4f*)(xg + (size_t)r * FEAT + c);
            *(v4f*)(&sX[r * XS + c]) = vx;

            // ---- stage Q row-fragment into swizzled B layout ----
            v4f vq = *(const v4f*)(qg + r * D + c);
            const int k   = r >> 2;               // K step (4 rows per step)
            const int p   = r & 1;                // position within the float2 pair
            const int hlf = (r >> 1) & 1;         // destination lane half
#pragma unroll
            for (int u = 0; u < 4; ++u) {
                const int e    = c + u;           // Q column
                const int t    = e >> 4;          // N tile
                const int c16v = e & 15;
                const int sig  = ((c16v << 1) + hlf + t) & 31;
                sB[(t * 32 + k) * 64 + sig * 2 + p] = vq[u];
            }
        }
    }
    __syncthreads();

    const int wave = tid >> 5;
    const int lane = tid & 31;
    const int half = lane >> 4;
    const int c16  = lane & 15;

    // ---- preload all 32 A fragments for this wave's 16 rows ----
    // A 16x4 f32 layout: lanes 0-15 hold (M=c16, K=4k+{0,1}); lanes 16-31 K=4k+{2,3}
    v2f a[32];
    {
        const float* ax = &sX[(wave * 16 + c16) * XS + 2 * half];
#pragma unroll
        for (int k = 0; k < 32; ++k)
            a[k] = *(const v2f*)(ax + 4 * k);     // ds_load_2addr_b64 pairs
    }

    float ss[8];
#pragma unroll
    for (int j = 0; j < 8; ++j) ss[j] = 0.0f;

    const int sig_base = (c16 << 1) + half;       // + t, & 31 per tile

    for (int t = 0; t < 8; t += 2) {              // 2 N-tiles per iteration
        const float* b0p = &sB[(t    ) * 2048 + (((sig_base + t    ) & 31) << 1)];
        const float* b1p = &sB[(t + 1) * 2048 + (((sig_base + t + 1) & 31) << 1)];
        v8f acc0 = {}, acc1 = {};
#pragma unroll
        for (int k = 0; k < 32; ++k) {
            v2f b0 = *(const v2f*)(b0p + k * 64); // ds_load_b64, imm offset k*256 B
            v2f b1 = *(const v2f*)(b1p + k * 64);
            acc0 = __builtin_amdgcn_wmma_f32_16x16x4_f32(
                false, a[k], false, b0, (short)0, acc0, false, false);
            acc1 = __builtin_amdgcn_wmma_f32_16x16x4_f32(
                false, a[k], false, b1, (short)0, acc1, false, false);
        }
        // C layout: VGPR j holds M = j + 8*half, N = c16 (within tile)
        float* zg0 = z + (size_t)(row0g + wave * 16 + half * 8) * FEAT
                       + (size_t)n * D + (t    ) * 16 + c16;
        float* zg1 = zg0 + 16;
#pragma unroll
        for (int j = 0; j < 8; ++j) {
            float v0 = acc0[j];
            float v1 = acc1[j];
            v0 = v0 > 0.0f ? v0 : 0.0f;
            v1 = v1 > 0.0f ? v1 : 0.0f;
            zg0[(size_t)j * FEAT] = v0;
            zg1[(size_t)j * FEAT] = v1;
            ss[j] += v0 * v0 + v1 * v1;
        }
    }

    // ---- per-row sum-of-squares: reduce across the 16 lanes of each half ----
#pragma unroll
    for (int j = 0; j < 8; ++j) {
        float v = ss[j];
        v += __shfl_xor(v, 8, 32);
        v += __shfl_xor(v, 4, 32);
        v += __shfl_xor(v, 2, 32);
        v += __shfl_xor(v, 1, 32);
        if (c16 == 0) {
            const int row = row0g + wave * 16 + half * 8 + j;
            atomicAdd(&sumsq[row], v);            // global_atomic_add_f32
        }
    }
}

// ---------------- pass 2: row-wise L2 normalize in place ----------------
__global__ void __launch_bounds__(256) brm_normalize(
    float* __restrict__ z, const float* __restrict__ sumsq)
{
    const size_t v4 = (size_t)blockIdx.x * 256 + threadIdx.x;  // float4 index
    const int row = (int)(v4 >> 10);                           // 1024 float4 per row
    const float s = sumsq[row];
    const float inv = 1.0f / fmaxf(sqrtf(s), 1e-12f);
    float* p = z + v4 * 4;
    v4f d = *(v4f*)p;
    d.x *= inv; d.y *= inv; d.z *= inv; d.w *= inv;
    *(v4f*)p = d;
}

// ---------------- launcher ----------------
extern "C" void kernel_launch(void* const* d_in, const int* in_sizes, int n_in,
                              void* d_out, int out_size, void* d_ws, size_t ws_size,
                              hipStream_t stream) {
    const float* x = (const float*)d_in[0];   // [16384, 4096] f32
    const float* Q = (const float*)d_in[1];   // [32, 128, 128] f32
    float* out   = (float*)d_out;             // [16384, 4096] f32
    float* sumsq = (float*)d_ws;              // [16384] f32 scratch

    brm_zero_sumsq<<<BATCH / 256, 256, 0, stream>>>(sumsq);

    dim3 grid(BATCH / ROWS_WG, NBLK);
    brm_block_gemm_relu<<<grid, 256, 0, stream>>>(x, Q, out, sumsq);

    const int nvec4 = (BATCH * FEAT) / 4;     // 16,777,216 float4
    brm_normalize<<<nvec4 / 256, 256, 0, stream>>>(out, sumsq);
}